// GCN_37744172597908
// MI455X (gfx1250) — compile-verified
//
#include <hip/hip_runtime.h>
#include <hip/hip_bf16.h>

typedef __attribute__((ext_vector_type(2))) float v2f;
typedef __attribute__((ext_vector_type(8))) float v8f;

#define HID 128
#define LDS_STRIDE 132   // 128 + 4 pad: rows land on distinct banks for frag reads

// ---------------- utility kernels ----------------

__global__ void fill_f32(float* __restrict__ p, float v, size_t n) {
    size_t i = (size_t)blockIdx.x * blockDim.x + threadIdx.x;
    if (i < n) p[i] = v;
}

__global__ void deg_scatter(const long long* __restrict__ dst, float* __restrict__ deg, int E) {
    int e = blockIdx.x * blockDim.x + threadIdx.x;
    if (e < E) atomicAdd(&deg[dst[e]], 1.0f);
}

__global__ void rsqrt_inplace(float* __restrict__ p, int n) {
    int i = blockIdx.x * blockDim.x + threadIdx.x;
    if (i < n) p[i] = rsqrtf(fmaxf(p[i], 1.0f));
}

// ---------------- WMMA f32 GEMM: Y[M,128] = X[M,128] @ W[128,128] ----------------
// grid.x = ceil(M/16) tiles; block = 256 (8 waves); wave w computes n-tile w.
__global__ __launch_bounds__(256) void gemm_wmma_f32(const float* __restrict__ X,
                                                     const float* __restrict__ W,
                                                     float* __restrict__ Y, int M) {
    __shared__ float tileA[16 * LDS_STRIDE];
    const int mBase = blockIdx.x * 16;
    const int tid = threadIdx.x;

    // cooperative, coalesced load of the 16x128 X tile into LDS (8 floats/thread)
    {
        int row = tid >> 4;          // 0..15
        int col = (tid & 15) * 8;    // 0..120
        int gr  = mBase + row; if (gr >= M) gr = M - 1;
        const float4* src = (const float4*)(X + (size_t)gr * HID + col);
        float4 a0 = src[0];
        float4 a1 = src[1];
        float* dp = &tileA[row * LDS_STRIDE + col];
        *(float4*)(dp)     = a0;
        *(float4*)(dp + 4) = a1;
    }
    __syncthreads();

    const int wave  = tid >> 5;       // 0..7 -> output column tile
    const int lane  = tid & 31;
    const int lo    = lane & 15;      // M (for A) / N (for B,C)
    const int hi    = lane >> 4;      // K-half select
    const int nBase = wave * 16;

    v8f c = {};
    const float* Arow = &tileA[lo * LDS_STRIDE + 2 * hi]; // K = k0 + v + 2*hi
    const float* Wc   = W + nBase + lo;                   // column lo of n-tile

    #pragma unroll 4
    for (int k0 = 0; k0 < HID; k0 += 4) {
        v2f a, b;
        a.x = Arow[k0];                                  // A[lo][k0 + 0 + 2*hi]
        a.y = Arow[k0 + 1];                              // A[lo][k0 + 1 + 2*hi]
        b.x = Wc[(size_t)(k0 + 2 * hi) * HID];           // W[k0 + 0 + 2*hi][nBase+lo]
        b.y = Wc[(size_t)(k0 + 1 + 2 * hi) * HID];       // W[k0 + 1 + 2*hi][nBase+lo]
        c = __builtin_amdgcn_wmma_f32_16x16x4_f32(false, a, false, b,
                                                  (short)0, c, false, false);
    }

    // C/D layout: VGPR p -> row (p + 8*hi), col lo
    float* Yb = Y + (size_t)(mBase + 8 * hi) * HID + nBase + lo;
    if (mBase + 16 <= M) {
        // full tile (always taken when M % 16 == 0): straight-line coalesced stores
        #pragma unroll
        for (int p = 0; p < 8; ++p) {
            Yb[(size_t)p * HID] = c[p];
        }
    } else {
        #pragma unroll
        for (int p = 0; p < 8; ++p) {
            int r = mBase + 8 * hi + p;
            if (r < M) Yb[(size_t)p * HID] = c[p];
        }
    }
}

// ---------------- edge aggregation: agg[dst] += h[src] * dinv[src]*dinv[dst] ----------------
// one thread per (edge, 4 features)
__global__ void edge_scatter(const float* __restrict__ H,
                             const long long* __restrict__ src,
                             const long long* __restrict__ dst,
                             const float* __restrict__ dinv,
                             float* __restrict__ agg, int E) {
    int t = blockIdx.x * blockDim.x + threadIdx.x;
    int e = t >> 5;
    if (e >= E) return;
    int c = (t & 31) * 4;
    long long s = src[e];
    long long d = dst[e];
    float nrm = dinv[s] * dinv[d];
    float4 h4 = *(const float4*)(H + (size_t)s * HID + c);
    float* ap = agg + (size_t)d * HID + c;
    atomicAdd(ap + 0, h4.x * nrm);
    atomicAdd(ap + 1, h4.y * nrm);
    atomicAdd(ap + 2, h4.z * nrm);
    atomicAdd(ap + 3, h4.w * nrm);
}

// h = relu(agg + h * dinv^2 + b)   (self-loop fused; in-place on h)
__global__ void bias_relu_self(float* __restrict__ h,
                               const float* __restrict__ agg,
                               const float* __restrict__ bias,
                               const float* __restrict__ dinv, int N) {
    int t = blockIdx.x * blockDim.x + threadIdx.x;
    int i = t >> 5;
    if (i >= N) return;
    int c = (t & 31) * 4;
    float di = dinv[i];
    float sl = di * di;
    float4 hv = *(const float4*)(h + (size_t)i * HID + c);
    float4 av = *(const float4*)(agg + (size_t)i * HID + c);
    float4 bv = *(const float4*)(bias + c);
    float4 r;
    r.x = fmaxf(av.x + hv.x * sl + bv.x, 0.0f);
    r.y = fmaxf(av.y + hv.y * sl + bv.y, 0.0f);
    r.z = fmaxf(av.z + hv.z * sl + bv.z, 0.0f);
    r.w = fmaxf(av.w + hv.w * sl + bv.w, 0.0f);
    *(float4*)(h + (size_t)i * HID + c) = r;
}

// out[i] = sum_c relu(agg[i][c] + h2[i][c]*dinv^2 + b2[c]) * Wr[c] + br
// one wave per node (wave32): lane handles 4 features, shfl reduction
__global__ __launch_bounds__(256) void head_kernel(const float* __restrict__ agg,
                                                   const float* __restrict__ h2,
                                                   const float* __restrict__ dinv,
                                                   const float* __restrict__ b2,
                                                   const float* __restrict__ Wr,
                                                   const float* __restrict__ br,
                                                   float* __restrict__ out, int N) {
    int wave = threadIdx.x >> 5;
    int lane = threadIdx.x & 31;
    int i = blockIdx.x * 8 + wave;
    if (i >= N) return;
    int c = lane * 4;
    float di = dinv[i];
    float sl = di * di;
    float4 av = *(const float4*)(agg + (size_t)i * HID + c);
    float4 hv = *(const float4*)(h2 + (size_t)i * HID + c);
    float4 bv = *(const float4*)(b2 + c);
    float4 wv = *(const float4*)(Wr + c);
    float acc = fmaxf(av.x + hv.x * sl + bv.x, 0.0f) * wv.x
              + fmaxf(av.y + hv.y * sl + bv.y, 0.0f) * wv.y
              + fmaxf(av.z + hv.z * sl + bv.z, 0.0f) * wv.z
              + fmaxf(av.w + hv.w * sl + bv.w, 0.0f) * wv.w;
    #pragma unroll
    for (int off = 16; off > 0; off >>= 1)
        acc += __shfl_xor(acc, off, 32);
    if (lane == 0) out[i] = acc + br[0];
}

// ---------------- launcher ----------------

extern "C" void kernel_launch(void* const* d_in, const int* in_sizes, int n_in,
                              void* d_out, int out_size, void* d_ws, size_t ws_size,
                              hipStream_t stream) {
    const float*     x  = (const float*)d_in[0];
    const long long* ei = (const long long*)d_in[1];   // int64 [2, E]
    const float*     W1 = (const float*)d_in[2];
    const float*     b1 = (const float*)d_in[3];
    const float*     W2 = (const float*)d_in[4];
    const float*     b2 = (const float*)d_in[5];
    const float*     Wr = (const float*)d_in[6];
    const float*     br = (const float*)d_in[7];
    float* out = (float*)d_out;

    const int N = in_sizes[0] / HID;   // 50000
    const int E = in_sizes[1] / 2;     // 600000
    const long long* srcIdx = ei;
    const long long* dstIdx = ei + E;

    // workspace layout
    float* dinv = (float*)d_ws;
    size_t off = (((size_t)N * sizeof(float)) + 255) & ~(size_t)255;
    float* A = (float*)((char*)d_ws + off);                            // N x 128
    float* B = (float*)((char*)d_ws + off + (size_t)N * HID * sizeof(float));

    const size_t NF = (size_t)N * HID;
    const int TB = 256;
    dim3 blk(TB);

    // degree -> 1/sqrt(deg)   (self-loop contributes the initial 1.0)
    fill_f32<<<dim3((N + TB - 1) / TB), blk, 0, stream>>>(dinv, 1.0f, (size_t)N);
    deg_scatter<<<dim3((E + TB - 1) / TB), blk, 0, stream>>>(dstIdx, dinv, E);
    rsqrt_inplace<<<dim3((N + TB - 1) / TB), blk, 0, stream>>>(dinv, N);

    // layer 1: h = x @ W1 ; agg ; h1' = relu(agg + self + b1)
    gemm_wmma_f32<<<dim3((N + 15) / 16), blk, 0, stream>>>(x, W1, A, N);
    fill_f32<<<dim3((unsigned)((NF + TB - 1) / TB)), blk, 0, stream>>>(B, 0.0f, NF);
    edge_scatter<<<dim3((unsigned)(((size_t)E * 32 + TB - 1) / TB)), blk, 0, stream>>>(
        A, srcIdx, dstIdx, dinv, B, E);
    bias_relu_self<<<dim3((unsigned)(((size_t)N * 32 + TB - 1) / TB)), blk, 0, stream>>>(
        A, B, b1, dinv, N);

    // layer 2: h2 = h1' @ W2 ; agg ; head fuses relu(agg + self + b2) . Wr + br
    gemm_wmma_f32<<<dim3((N + 15) / 16), blk, 0, stream>>>(A, W2, B, N);
    fill_f32<<<dim3((unsigned)((NF + TB - 1) / TB)), blk, 0, stream>>>(A, 0.0f, NF);
    edge_scatter<<<dim3((unsigned)(((size_t)E * 32 + TB - 1) / TB)), blk, 0, stream>>>(
        B, srcIdx, dstIdx, dinv, A, E);
    head_kernel<<<dim3((N + 7) / 8), blk, 0, stream>>>(A, B, dinv, b2, Wr, br, out, N);
}